// GCNEncoder_4363686773056
// MI455X (gfx1250) — compile-verified
//
#include <hip/hip_runtime.h>

typedef __attribute__((ext_vector_type(2))) float v2f;
typedef __attribute__((ext_vector_type(8))) float v8f;

// ---------------------------------------------------------------------------
// Degree / normalization kernels
// ---------------------------------------------------------------------------
__global__ void k_fill1(float* __restrict__ p, int n) {
    int i = blockIdx.x * blockDim.x + threadIdx.x;
    if (i < n) p[i] = 1.0f;   // self-loop contributes 1 to every node's degree
}

__global__ void k_degcount(const int* __restrict__ dst, float* __restrict__ deg, int nE) {
    int e = blockIdx.x * blockDim.x + threadIdx.x;
    if (e < nE) atomicAdd(&deg[dst[e]], 1.0f);
}

__global__ void k_rsqrt_inplace(float* __restrict__ p, int n) {
    int i = blockIdx.x * blockDim.x + threadIdx.x;
    if (i < n) p[i] = rsqrtf(p[i]);   // deg >= 1 always (self loop)
}

// ---------------------------------------------------------------------------
// FP32 WMMA GEMM:  C[M x N] = op(A)[M x K] @ B[K x N],  N = NT*16, M % 16 == 0
// One wave computes a 16 x N row-tile using V_WMMA_F32_16X16X4_F32.
// A-frag layout (16x4): lane<16 -> K={kk,kk+1}, lane>=16 -> K={kk+2,kk+3}
// B-frag layout (4x16): VGPR0 = rows kk / kk+2, VGPR1 = rows kk+1 / kk+3
// C/D layout: VGPR r -> (M=r, lanes 0-15) and (M=r+8, lanes 16-31)
// ---------------------------------------------------------------------------
template <int K, int NT, bool RELU_A>
__global__ __launch_bounds__(256) void k_gemm_wmma(const float* __restrict__ A,
                                                   const float* __restrict__ B,
                                                   float* __restrict__ C, int M) {
    constexpr int N = NT * 16;
    const int lane = threadIdx.x & 31;
    const int wave = threadIdx.x >> 5;
    const int wavesPerBlock = blockDim.x >> 5;
    const long long row0 = (long long)(blockIdx.x * wavesPerBlock + wave) * 16;
    if (row0 >= M) return;                 // wave-uniform branch: EXEC stays all-ones

    const int half = lane >> 4;            // 0: K pair {kk,kk+1}, 1: {kk+2,kk+3}
    const int l15  = lane & 15;

    v8f acc[NT] = {};

    const long long arow = row0 + l15;
    const float* __restrict__ Arow = A + arow * K;

    for (int kk = 0; kk < K; kk += 4) {
        const int ka = kk + half * 2;
        v2f a = *reinterpret_cast<const v2f*>(Arow + ka);   // 8B aligned (ka even, K even)
        if (RELU_A) {
            a.x = fmaxf(a.x, 0.0f);
            a.y = fmaxf(a.y, 0.0f);
        }
#pragma unroll
        for (int t = 0; t < NT; ++t) {
            const int col = t * 16 + l15;
            v2f b;
            b.x = B[(long long)ka * N + col];
            b.y = B[(long long)(ka + 1) * N + col];
            acc[t] = __builtin_amdgcn_wmma_f32_16x16x4_f32(
                false, a, false, b, (short)0, acc[t], false, false);
        }
    }

#pragma unroll
    for (int t = 0; t < NT; ++t) {
        const int col = t * 16 + l15;
        const long long rbase = row0 + half * 8;
#pragma unroll
        for (int r = 0; r < 8; ++r) {
            C[(rbase + r) * N + col] = acc[t][r];
        }
    }
}

// ---------------------------------------------------------------------------
// Aggregation: out[i][c] = bias[c] + dinv[i]^2 * h[i][c]   (self-loop + bias)
// ---------------------------------------------------------------------------
template <int F, int LOGF>
__global__ void k_agg_init(const float* __restrict__ h, const float* __restrict__ dinv,
                           const float* __restrict__ bias, float* __restrict__ out,
                           int nNodes) {
    const long long t = (long long)blockIdx.x * blockDim.x + threadIdx.x;
    if (t >= ((long long)nNodes << LOGF)) return;
    const int i = (int)(t >> LOGF);
    const int c = (int)(t & (F - 1));
    const float di = dinv[i];
    out[t] = bias[c] + di * di * h[t];
}

// ---------------------------------------------------------------------------
// Edge scatter: out[dst][c] += dinv[src]*dinv[dst] * h[src][c]
// F consecutive lanes share one edge -> dinv loads broadcast, h/out coalesced.
// ---------------------------------------------------------------------------
template <int F, int LOGF>
__global__ void k_agg_edges(const float* __restrict__ h, const float* __restrict__ dinv,
                            const int* __restrict__ src, const int* __restrict__ dst,
                            float* __restrict__ out, int nE) {
    const long long t = (long long)blockIdx.x * blockDim.x + threadIdx.x;
    const long long e = t >> LOGF;
    if (e >= nE) return;
    const int c = (int)(t & (F - 1));
    const int s = src[e];
    const int d = dst[e];
    const float w = dinv[s] * dinv[d];
    atomicAdd(&out[((long long)d << LOGF) + c], w * h[((long long)s << LOGF) + c]);
}

// ---------------------------------------------------------------------------
// Launcher
// ---------------------------------------------------------------------------
extern "C" void kernel_launch(void* const* d_in, const int* in_sizes, int n_in,
                              void* d_out, int out_size, void* d_ws, size_t ws_size,
                              hipStream_t stream) {
    constexpr int IN_CH  = 128;
    constexpr int HID    = 64;
    constexpr int OUT_CH = 32;

    const float* x   = (const float*)d_in[0];
    const int*   ei  = (const int*)d_in[1];
    const float* W1  = (const float*)d_in[2];
    const float* b1  = (const float*)d_in[3];
    const float* W2  = (const float*)d_in[4];
    const float* b2  = (const float*)d_in[5];
    float*       out = (float*)d_out;

    const int nNodes = in_sizes[0] / IN_CH;     // 80000
    const int nE     = in_sizes[1] / 2;         // 1280000
    const int* src = ei;
    const int* dst = ei + nE;

    // Workspace carve-up (256B aligned blocks)
    char* ws = (char*)d_ws;
    const size_t dinvB = ((size_t)nNodes * 4 + 255) & ~(size_t)255;
    const size_t h1B   = ((size_t)nNodes * HID * 4 + 255) & ~(size_t)255;
    const size_t a1B   = h1B;
    float* dinv = (float*)ws;
    float* h1   = (float*)(ws + dinvB);
    float* a1   = (float*)(ws + dinvB + h1B);
    float* h2   = (float*)(ws + dinvB + h1B + a1B);
    (void)ws_size; (void)n_in; (void)out_size;

    const int T = 256;
    auto blocks = [](long long n, int t) { return (unsigned)((n + t - 1) / t); };

    // 1) symmetric normalization:  dinv = rsqrt(1 + in_degree)
    k_fill1<<<blocks(nNodes, T), T, 0, stream>>>(dinv, nNodes);
    k_degcount<<<blocks(nE, T), T, 0, stream>>>(dst, dinv, nE);
    k_rsqrt_inplace<<<blocks(nNodes, T), T, 0, stream>>>(dinv, nNodes);

    // 2) h1 = x @ W1   [N x 128] @ [128 x 64]   (8 waves/block, 16 rows/wave)
    k_gemm_wmma<IN_CH, HID / 16, false>
        <<<blocks((long long)nNodes / 128, 1), T, 0, stream>>>(x, W1, h1, nNodes);

    // 3) a1 = b1 + dinv^2*h1 (self loop) then scatter edges
    k_agg_init<HID, 6><<<blocks((long long)nNodes * HID, T), T, 0, stream>>>(
        h1, dinv, b1, a1, nNodes);
    k_agg_edges<HID, 6><<<blocks((long long)nE * HID, T), T, 0, stream>>>(
        h1, dinv, src, dst, a1, nE);

    // 4) h2 = relu(a1) @ W2   [N x 64] @ [64 x 32]  (relu fused into A loads)
    k_gemm_wmma<HID, OUT_CH / 16, true>
        <<<blocks((long long)nNodes / 128, 1), T, 0, stream>>>(a1, W2, h2, nNodes);

    // 5) out = b2 + dinv^2*h2 (self loop) then scatter edges
    k_agg_init<OUT_CH, 5><<<blocks((long long)nNodes * OUT_CH, T), T, 0, stream>>>(
        h2, dinv, b2, out, nNodes);
    k_agg_edges<OUT_CH, 5><<<blocks((long long)nE * OUT_CH, T), T, 0, stream>>>(
        h2, dinv, src, dst, out, nE);
}